// GCN_3255585210425
// MI455X (gfx1250) — compile-verified
//
#include <hip/hip_runtime.h>
#include <math.h>

typedef __attribute__((ext_vector_type(2))) float v2f;
typedef __attribute__((ext_vector_type(8))) float v8f;

#define N_IN   165
#define N_HID  64
#define N_OUT  2
#define K_PAD  168          // K rounded up to multiple of 4 (zero-padded in LDS)
#define LDS_STRIDE 170      // row stride in floats: even (8B-aligned pairs), 42 mod 64 -> conflict-free

// ---------------------------------------------------------------- utilities
__global__ void fill_f32(float* __restrict__ p, long long n, float v) {
    long long i = (long long)blockIdx.x * blockDim.x + threadIdx.x;
    if (i < n) p[i] = v;
}

// deg[dst] += 1 per edge (deg pre-initialized to 1.0 for the self-loop)
__global__ void degree_kernel(const int* __restrict__ dst, float* __restrict__ deg, int n_edges) {
    int e = blockIdx.x * blockDim.x + threadIdx.x;
    if (e < n_edges) atomicAdd(&deg[dst[e]], 1.0f);
}

// deg -> rsqrt(deg) in place
__global__ void rsqrt_kernel(float* __restrict__ deg, int n) {
    int i = blockIdx.x * blockDim.x + threadIdx.x;
    if (i < n) deg[i] = rsqrtf(deg[i]);
}

// --------------------------------------------------- layer 1 GEMM via WMMA
// h1[N,64] = x[N,165] @ W1[165,64], fp32 WMMA 16x16x4.
// Block = 256 threads = 8 waves, covers 32 rows x 64 cols.
// Both operands staged in LDS, zero-padded to K=168 -> branch-free inner loop.
__global__ void __launch_bounds__(256)
gemm1_wmma(const float* __restrict__ x, const float* __restrict__ W,
           float* __restrict__ h, int n_nodes) {
    __shared__ float smem[(32 + 64) * LDS_STRIDE];
    float* Xs = smem;                       // [32][LDS_STRIDE] : x tile
    float* Wt = smem + 32 * LDS_STRIDE;     // [64][LDS_STRIDE] : W1 transposed

    const int tid     = threadIdx.x;
    const int rowbase = blockIdx.x * 32;

    // stage x tile: linear index maps to contiguous global addresses (coalesced)
    for (int idx = tid; idx < 32 * N_IN; idx += 256) {
        int r = idx / N_IN;
        int k = idx - r * N_IN;
        int row = rowbase + r;
        float v = 0.f;
        if (row < n_nodes) v = x[(size_t)row * N_IN + k];
        Xs[r * LDS_STRIDE + k] = v;
    }
    // zero-pad x columns 165..169
    for (int idx = tid; idx < 32 * 5; idx += 256) {
        int r = idx / 5;
        int k = N_IN + (idx - r * 5);
        Xs[r * LDS_STRIDE + k] = 0.f;
    }
    // stage W1 transposed: global reads coalesced (W row-major, 64-wide)
    for (int idx = tid; idx < N_IN * N_HID; idx += 256) {
        int k = idx >> 6;       // 0..164
        int n = idx & 63;
        Wt[n * LDS_STRIDE + k] = W[idx];
    }
    // zero-pad W columns 165..169
    for (int idx = tid; idx < 5 * N_HID; idx += 256) {
        int k = N_IN + (idx >> 6);
        int n = idx & 63;
        Wt[n * LDS_STRIDE + k] = 0.f;
    }
    __syncthreads();

    const int wave = tid >> 5;
    const int lane = tid & 31;
    const int half = lane >> 4;                 // 0: K pair {k,k+1}; 1: {k+2,k+3}
    const int l16  = lane & 15;
    const int rl   = (wave & 1) * 16 + l16;     // local A row (M = l16 for all lanes)
    const int bc   = (wave >> 1) * 16 + l16;    // B column (N = l16)

    const float* ax = Xs + rl * LDS_STRIDE + 2 * half;
    const float* bw = Wt + bc * LDS_STRIDE + 2 * half;

    v8f acc = {};
    #pragma unroll 6
    for (int k = 0; k < K_PAD; k += 4) {
        v2f a = *(const v2f*)(ax + k);          // ds_load_2addr_b64 (8B aligned)
        v2f b = *(const v2f*)(bw + k);
        acc = __builtin_amdgcn_wmma_f32_16x16x4_f32(
            /*neg_a=*/false, a, /*neg_b=*/false, b,
            /*c_mod=*/(short)0, acc, /*reuse_a=*/false, /*reuse_b=*/false);
    }

    // C/D layout: VGPR r holds M = r + half*8, N = l16.
    // Tile bound is wave-uniform -> one scalar branch; fast path is 8
    // unconditional coalesced stores.
    const int row0 = rowbase + (wave & 1) * 16;
    if (row0 + 16 <= n_nodes) {
        float* hp = h + (size_t)(row0 + half * 8) * N_HID + bc;
        #pragma unroll
        for (int r = 0; r < 8; ++r)
            hp[(size_t)r * N_HID] = acc[r];
    } else {
        #pragma unroll
        for (int r = 0; r < 8; ++r) {
            int row = row0 + r + half * 8;
            if (row < n_nodes) h[(size_t)row * N_HID + bc] = acc[r];
        }
    }
}

// ----------------------------------------------- layer 1 edge scatter (64ch)
// 64 threads per edge: coalesced gather of h1[src] row, atomic add into agg[dst]
__global__ void scatter1_kernel(const int* __restrict__ src, const int* __restrict__ dst,
                                const float* __restrict__ dinv,
                                const float* __restrict__ h1,
                                float* __restrict__ agg, int n_edges) {
    long long gid = (long long)blockIdx.x * blockDim.x + threadIdx.x;
    int e  = (int)(gid >> 6);
    int ch = (int)(gid & 63);
    if (e >= n_edges) return;
    int s = src[e], d = dst[e];
    float c = dinv[s] * dinv[d];
    atomicAdd(&agg[(size_t)d * N_HID + ch], h1[(size_t)s * N_HID + ch] * c);
}

// self-loop + bias + ReLU, result written back into h1 buffer
__global__ void self1_kernel(const float* __restrict__ agg, float* __restrict__ h1,
                             const float* __restrict__ dinv, const float* __restrict__ b1,
                             int n_nodes) {
    long long gid = (long long)blockIdx.x * blockDim.x + threadIdx.x;
    int node = (int)(gid >> 6);
    int ch   = (int)(gid & 63);
    if (node >= n_nodes) return;
    float di = dinv[node];
    float v = agg[gid] + h1[gid] * (di * di) + b1[ch];
    h1[gid] = v > 0.f ? v : 0.f;
}

// --------------------------------------------------- layer 2 (64 -> 2) GEMM
__global__ void gemm2_kernel(const float* __restrict__ h1, const float* __restrict__ W2,
                             float* __restrict__ h2, int n_nodes) {
    int i = blockIdx.x * blockDim.x + threadIdx.x;
    if (i >= n_nodes) return;
    const float* r = h1 + (size_t)i * N_HID;
    float s0 = 0.f, s1 = 0.f;
    #pragma unroll 8
    for (int k = 0; k < N_HID; ++k) {
        float v = r[k];
        s0 = fmaf(v, W2[k * N_OUT + 0], s0);
        s1 = fmaf(v, W2[k * N_OUT + 1], s1);
    }
    h2[(size_t)i * N_OUT + 0] = s0;
    h2[(size_t)i * N_OUT + 1] = s1;
}

// ------------------------------------------------ layer 2 edge scatter (2ch)
__global__ void scatter2_kernel(const int* __restrict__ src, const int* __restrict__ dst,
                                const float* __restrict__ dinv,
                                const float* __restrict__ h2,
                                float* __restrict__ out, int n_edges) {
    int e = blockIdx.x * blockDim.x + threadIdx.x;
    if (e >= n_edges) return;
    int s = src[e], d = dst[e];
    float c = dinv[s] * dinv[d];
    atomicAdd(&out[(size_t)d * N_OUT + 0], h2[(size_t)s * N_OUT + 0] * c);
    atomicAdd(&out[(size_t)d * N_OUT + 1], h2[(size_t)s * N_OUT + 1] * c);
}

// self-loop + bias for layer 2
__global__ void final_kernel(const float* __restrict__ h2, const float* __restrict__ dinv,
                             const float* __restrict__ b2, float* __restrict__ out,
                             int n_nodes) {
    int i = blockIdx.x * blockDim.x + threadIdx.x;
    if (i >= n_nodes) return;
    float di = dinv[i];
    float d2 = di * di;
    out[(size_t)i * N_OUT + 0] += h2[(size_t)i * N_OUT + 0] * d2 + b2[0];
    out[(size_t)i * N_OUT + 1] += h2[(size_t)i * N_OUT + 1] * d2 + b2[1];
}

extern "C" void kernel_launch(void* const* d_in, const int* in_sizes, int n_in,
                              void* d_out, int out_size, void* d_ws, size_t ws_size,
                              hipStream_t stream) {
    const float* x  = (const float*)d_in[0];
    const int*   ei = (const int*)  d_in[1];
    const float* W1 = (const float*)d_in[2];
    const float* b1 = (const float*)d_in[3];
    const float* W2 = (const float*)d_in[4];
    const float* b2 = (const float*)d_in[5];
    float* out = (float*)d_out;

    const int N = in_sizes[0] / N_IN;   // 200000
    const int E = in_sizes[1] / 2;      // 1000000
    const int* src = ei;
    const int* dst = ei + E;

    // workspace layout (fp32): dinv[N] | h1[N*64] | agg[N*64] | h2[N*2]
    float* dinv = (float*)d_ws;
    float* h1   = dinv + N;
    float* agg  = h1 + (size_t)N * N_HID;
    float* h2   = agg + (size_t)N * N_HID;

    const int T = 256;
    long long nh = (long long)N * N_HID;

    // init: deg = 1 (self-loop), agg = 0, out = 0
    fill_f32<<<(unsigned)((N + T - 1) / T), T, 0, stream>>>(dinv, N, 1.0f);
    fill_f32<<<(unsigned)((nh + T - 1) / T), T, 0, stream>>>(agg, nh, 0.0f);
    fill_f32<<<(unsigned)(((long long)N * N_OUT + T - 1) / T), T, 0, stream>>>(out, (long long)N * N_OUT, 0.0f);

    // degree + dinv
    degree_kernel<<<(E + T - 1) / T, T, 0, stream>>>(dst, dinv, E);
    rsqrt_kernel<<<(N + T - 1) / T, T, 0, stream>>>(dinv, N);

    // layer 1
    gemm1_wmma<<<(N + 31) / 32, T, 0, stream>>>(x, W1, h1, N);
    long long s1 = (long long)E * N_HID;
    scatter1_kernel<<<(unsigned)((s1 + T - 1) / T), T, 0, stream>>>(src, dst, dinv, h1, agg, E);
    self1_kernel<<<(unsigned)((nh + T - 1) / T), T, 0, stream>>>(agg, h1, dinv, b1, N);

    // layer 2
    gemm2_kernel<<<(N + T - 1) / T, T, 0, stream>>>(h1, W2, h2, N);
    scatter2_kernel<<<(E + T - 1) / T, T, 0, stream>>>(src, dst, dinv, h2, out, E);
    final_kernel<<<(N + T - 1) / T, T, 0, stream>>>(h2, dinv, b2, out, N);
}